// OnnxSafeHTDemucs_78984448573789
// MI455X (gfx1250) — compile-verified
//
#include <hip/hip_runtime.h>
#include <hip/hip_bf16.h>
#include <math.h>

// ---------------- problem constants (derived from the reference) ----------
#define T_LEN   441000          // input samples per channel
#define HOP     1024
#define NFFT    4096
#define NFREQ   2048            // frequencies actually used (row 2048 is zeroed)
#define PAD1    1536            // HOP/2*3
#define L1      444416          // 2*PAD1 + le*HOP, le = 431
#define P2      2048            // NFFT/2
#define LP      448512          // L1 + 2*P2  (padded signal length / T_out)
#define NF      435             // number of STFT frames
#define NFP     448             // frames padded to multiple of 64 for tiling
#define BC      16              // 8 batch * 2 channels
#define OUT_OFF 3584            // P2 + PAD1 : slice offset into reconstructed signal

typedef float v2f __attribute__((ext_vector_type(2)));
typedef float v8f __attribute__((ext_vector_type(8)));
typedef unsigned int u32x4 __attribute__((ext_vector_type(4)));
typedef int i32x4 __attribute__((ext_vector_type(4)));
typedef int i32x8 __attribute__((ext_vector_type(8)));

#if __has_builtin(__builtin_amdgcn_tensor_load_to_lds)
#define USE_TDM 1
#else
#define USE_TDM 0
#endif

// fp32 matrix-core op: D(16x16) = A(16x4) * B(4x16) + C, full fp32 precision.
__device__ __forceinline__ v8f wmma_f32(v2f a, v2f b, v8f c) {
    return __builtin_amdgcn_wmma_f32_16x16x4_f32(
        false, a, false, b, (short)0, c, false, false);
}

#if USE_TDM
// LDS byte offset of a __shared__ object (addrspace(3) ptrtoint).
__device__ __forceinline__ unsigned lds_offset(void* p) {
    return (unsigned)(unsigned long long)(__attribute__((address_space(3))) void*)p;
}

// Issue one TDM 2-D tile load: tensor row stride = stride0 (elements, 4B each),
// tile = tile_d0 (contiguous, X) x tile_d1 (rows, Y), zero-fill rows >= tensor_d1.
// LDS pitch = tile_d0 + pad DWORDs via pad_interval/pad_amount codes.
__device__ __forceinline__ void tdm_load_2d(unsigned lds_addr, const void* gptr,
                                            unsigned tile_d0, unsigned tile_d1,
                                            unsigned tensor_d0, unsigned tensor_d1,
                                            unsigned long long stride0,
                                            unsigned pad_int_code, unsigned pad_amt_code)
{
    const unsigned long long ga = (unsigned long long)gptr;
    u32x4 g0;
    g0[0] = 1u;                                   // count=1, user descriptor, no gather
    g0[1] = lds_addr;                             // LDS byte address
    g0[2] = (unsigned)ga;                         // global_addr[31:0]
    g0[3] = (unsigned)(ga >> 32) | 0x80000000u;   // global_addr[56:32] | type=2

    i32x8 g1;
    g1[0] = (int)((2u << 16)                      // data_size = 4 bytes
                | (1u << 20)                      // pad_enable
                | (pad_int_code << 22)            // DWORDs between pads
                | (pad_amt_code << 25));          // pad DWORDs - 1
    g1[1] = (int)((tensor_d0 & 0xFFFFu) << 16);                       // tensor_dim0[15:0]
    g1[2] = (int)((tensor_d0 >> 16) | ((tensor_d1 & 0xFFFFu) << 16)); // td0[31:16]|td1[15:0]
    g1[3] = (int)((tensor_d1 >> 16) | (tile_d0 << 16));               // td1[31:16]|tile_dim0
    g1[4] = (int)(tile_d1 & 0xFFFFu);                                 // tile_dim1 (tile_dim2=0)
    g1[5] = (int)(unsigned)(stride0 & 0xFFFFFFFFull);                 // dim0_stride[31:0]
    g1[6] = (int)(unsigned)((stride0 >> 32) & 0xFFFFull);             // dim0_stride[47:32]
    g1[7] = 0;

    const i32x4 z4 = {0, 0, 0, 0};                // 2-D tensor: groups 2/3 unused
    const i32x8 z8 = {0, 0, 0, 0, 0, 0, 0, 0};
    // 6-arg form (clang-23 / therock-10.0 headers): (g0, g1, g2, g3, extra, cpol)
    __builtin_amdgcn_tensor_load_to_lds(g0, g1, z4, z4, z8, 0);
}
#endif

// ---------------- kernel 1: double reflect pad -----------------------------
__global__ void pad_kernel(const float* __restrict__ mix, float* __restrict__ xp) {
    const int bc = blockIdx.y;
    const int j  = blockIdx.x * blockDim.x + threadIdx.x;
    if (j >= LP) return;
    int j2 = j - P2;
    if (j2 < 0)   j2 = -j2;
    if (j2 >= L1) j2 = 2 * L1 - 2 - j2;
    int j1 = j2 - PAD1;
    if (j1 < 0)      j1 = -j1;
    if (j1 >= T_LEN) j1 = 2 * T_LEN - 2 - j1;
    xp[(size_t)bc * LP + j] = mix[(size_t)bc * T_LEN + j1];
}

// ---------------- zero fill ------------------------------------------------
__global__ void zero_kernel(float* __restrict__ p, int n4) {
    int i = blockIdx.x * blockDim.x + threadIdx.x;
    if (i < n4) reinterpret_cast<float4*>(p)[i] = make_float4(0.f, 0.f, 0.f, 0.f);
}

// ---------------- kernel 2: STFT GEMM --------------------------------------
// r[k,t] = sum_n br[k,n]*xp[t*HOP+n];  i stored negated.
// M=freq(2048), N=frames(448), K=4096. WG tile 128x64, K chunks of 16,
// double-buffered LDS fed by the Tensor Data Mover (wave 0 drives the DMA).
__launch_bounds__(256, 2)
__global__ void stft_gemm(const float* __restrict__ xp,
                          const float* __restrict__ br,
                          const float* __restrict__ bi,
                          float* __restrict__ rbuf,
                          float* __restrict__ ibuf) {
    __shared__ float Ar[2][128][20];   // basis_real tile [m][k], pitch 20
    __shared__ float Ai[2][128][20];
    __shared__ float Bt[2][64][20];    // frame tile [t][k]

    const int bc   = blockIdx.z;
    const int m0   = blockIdx.x * 128;
    const int t0   = blockIdx.y * 64;
    const int tid  = threadIdx.x;
    const int lane = tid & 31;
    const int w    = tid >> 5;

    const float* xpc = xp + (size_t)bc * LP;

    v8f accr[4], acci[4];
#pragma unroll
    for (int q = 0; q < 4; ++q) { accr[q] = v8f{0,0,0,0,0,0,0,0}; acci[q] = v8f{0,0,0,0,0,0,0,0}; }

    const int NCH = NFFT / 16;   // 256 K-chunks

#if USE_TDM
    const unsigned brows = (unsigned)(NF - t0);           // zero-fill frames >= NF
    if (tid < 32) {                                       // prologue: chunk 0 -> buf 0
        tdm_load_2d(lds_offset(&Ar[0][0][0]), br + (size_t)m0 * NFFT, 16, 128, 16, 128, NFFT, 3, 3);
        tdm_load_2d(lds_offset(&Ai[0][0][0]), bi + (size_t)m0 * NFFT, 16, 128, 16, 128, NFFT, 3, 3);
        tdm_load_2d(lds_offset(&Bt[0][0][0]), xpc + (size_t)t0 * HOP, 16, 64, 16, brows, HOP, 3, 3);
    }
#endif

    for (int c = 0; c < NCH; ++c) {
#if USE_TDM
        const int buf = c & 1;
        if (tid < 32) {
            if (c + 1 < NCH) {                            // prefetch next chunk
                const int kn = (c + 1) * 16;
                tdm_load_2d(lds_offset(&Ar[buf ^ 1][0][0]), br + (size_t)m0 * NFFT + kn, 16, 128, 16, 128, NFFT, 3, 3);
                tdm_load_2d(lds_offset(&Ai[buf ^ 1][0][0]), bi + (size_t)m0 * NFFT + kn, 16, 128, 16, 128, NFFT, 3, 3);
                tdm_load_2d(lds_offset(&Bt[buf ^ 1][0][0]), xpc + (size_t)t0 * HOP + kn, 16, 64, 16, brows, HOP, 3, 3);
                __builtin_amdgcn_s_wait_tensorcnt(3);     // current chunk landed (in-order)
            } else {
                __builtin_amdgcn_s_wait_tensorcnt(0);
            }
        }
#else
        const int buf = 0;
        const int k0 = c * 16;
#pragma unroll
        for (int p = 0; p < 2; ++p) {                     // A tiles: 128x16 each
            const int lin = tid + p * 256;
            const int row = lin >> 2, cg = (lin & 3) << 2;
            *reinterpret_cast<float4*>(&Ar[0][row][cg]) =
                *reinterpret_cast<const float4*>(br + (size_t)(m0 + row) * NFFT + k0 + cg);
            *reinterpret_cast<float4*>(&Ai[0][row][cg]) =
                *reinterpret_cast<const float4*>(bi + (size_t)(m0 + row) * NFFT + k0 + cg);
        }
        {                                                 // B tile: 64x16
            const int tl = tid >> 2, cg = (tid & 3) << 2;
            const int t = t0 + tl;
            float4 bv = make_float4(0.f, 0.f, 0.f, 0.f);
            if (t < NF) bv = *reinterpret_cast<const float4*>(xpc + (size_t)t * HOP + k0 + cg);
            *reinterpret_cast<float4*>(&Bt[0][tl][cg]) = bv;
        }
#endif
        __syncthreads();

        const int mrow = (w << 4) + (lane & 15);
#pragma unroll
        for (int kk = 0; kk < 16; kk += 4) {
            const int kc = kk + ((lane >> 4) << 1);       // K = (lane>>4)*2 + j
            const v2f a_r = *reinterpret_cast<const v2f*>(&Ar[buf][mrow][kc]);
            const v2f a_i = *reinterpret_cast<const v2f*>(&Ai[buf][mrow][kc]);
#pragma unroll
            for (int nt = 0; nt < 4; ++nt) {
                const v2f b = *reinterpret_cast<const v2f*>(&Bt[buf][(nt << 4) + (lane & 15)][kc]);
                accr[nt] = wmma_f32(a_r, b, accr[nt]);
                acci[nt] = wmma_f32(a_i, b, acci[nt]);
            }
        }
        __syncthreads();
    }

    // epilogue: D layout lane L, vgpr v -> M = v + 8*(L>>4), N = L&15
    float* rout = rbuf + (size_t)bc * NFREQ * NFP;
    float* iout = ibuf + (size_t)bc * NFREQ * NFP;
    const int mbase = m0 + (w << 4) + ((lane >> 4) << 3);
    const int tcol0 = t0 + (lane & 15);
#pragma unroll
    for (int nt = 0; nt < 4; ++nt) {
        const int t = tcol0 + (nt << 4);
        const bool keep = (t >= 2) && (t < 433);          // frames {0,1,433,434+} zeroed
#pragma unroll
        for (int v = 0; v < 8; ++v) {
            const int k = mbase + v;
            rout[(size_t)k * NFP + t] = keep ?  accr[nt][v] : 0.0f;
            iout[(size_t)k * NFP + t] = keep ? -acci[nt][v] : 0.0f;   // pre-negate imag
        }
    }
}

// ---------------- kernel 3: ISTFT GEMM + overlap-add scatter ---------------
// out[t,n] = sum_k r[k,t]*br[k,n] + (-i)[k,t]*bi[k,n]  (syn*win == basis)
// M=t(448), N=n(4096), K=2048. WG tile 64x128, waves 4x2, TDM double buffer.
__launch_bounds__(256, 2)
__global__ void istft_gemm_scatter(const float* __restrict__ rbuf,
                                   const float* __restrict__ ibuf,
                                   const float* __restrict__ br,
                                   const float* __restrict__ bi,
                                   float* __restrict__ sig) {
    __shared__ float Alr[2][16][68];    // r tile  [k][t], pitch 68
    __shared__ float Ali[2][16][68];    // -i tile [k][t]
    __shared__ float Blr[2][16][132];   // br tile [k][n], pitch 132
    __shared__ float Bli[2][16][132];   // bi tile [k][n]

    const int bc   = blockIdx.z;
    const int t0   = blockIdx.x * 64;
    const int n0   = blockIdx.y * 128;
    const int tid  = threadIdx.x;
    const int lane = tid & 31;
    const int w    = tid >> 5;
    const int mw   = (w >> 1) << 4;     // 0,16,32,48
    const int nw   = (w & 1) << 6;      // 0,64

    const float* rc = rbuf + (size_t)bc * NFREQ * NFP;
    const float* ic = ibuf + (size_t)bc * NFREQ * NFP;

    v8f acc[4];
#pragma unroll
    for (int q = 0; q < 4; ++q) acc[q] = v8f{0,0,0,0,0,0,0,0};

    const int NCH = NFREQ / 16;         // 128 K-chunks

#if USE_TDM
    if (tid < 32) {                     // prologue: chunk 0 -> buf 0
        tdm_load_2d(lds_offset(&Alr[0][0][0]), rc + t0, 64, 16, 64, 16, NFP, 5, 3);
        tdm_load_2d(lds_offset(&Ali[0][0][0]), ic + t0, 64, 16, 64, 16, NFP, 5, 3);
        tdm_load_2d(lds_offset(&Blr[0][0][0]), br + n0, 128, 16, 128, 16, NFFT, 6, 3);
        tdm_load_2d(lds_offset(&Bli[0][0][0]), bi + n0, 128, 16, 128, 16, NFFT, 6, 3);
    }
#endif

    for (int c = 0; c < NCH; ++c) {
#if USE_TDM
        const int buf = c & 1;
        if (tid < 32) {
            if (c + 1 < NCH) {
                const size_t ka = (size_t)(c + 1) * 16 * NFP;
                const size_t kb = (size_t)(c + 1) * 16 * NFFT;
                tdm_load_2d(lds_offset(&Alr[buf ^ 1][0][0]), rc + ka + t0, 64, 16, 64, 16, NFP, 5, 3);
                tdm_load_2d(lds_offset(&Ali[buf ^ 1][0][0]), ic + ka + t0, 64, 16, 64, 16, NFP, 5, 3);
                tdm_load_2d(lds_offset(&Blr[buf ^ 1][0][0]), br + kb + n0, 128, 16, 128, 16, NFFT, 6, 3);
                tdm_load_2d(lds_offset(&Bli[buf ^ 1][0][0]), bi + kb + n0, 128, 16, 128, 16, NFFT, 6, 3);
                __builtin_amdgcn_s_wait_tensorcnt(4);
            } else {
                __builtin_amdgcn_s_wait_tensorcnt(0);
            }
        }
#else
        const int buf = 0;
        const int k0 = c * 16;
        {                               // A tiles: 16x64 each
            const int kl = tid >> 4, tg = (tid & 15) << 2;
            *reinterpret_cast<float4*>(&Alr[0][kl][tg]) =
                *reinterpret_cast<const float4*>(rc + (size_t)(k0 + kl) * NFP + t0 + tg);
            *reinterpret_cast<float4*>(&Ali[0][kl][tg]) =
                *reinterpret_cast<const float4*>(ic + (size_t)(k0 + kl) * NFP + t0 + tg);
        }
#pragma unroll
        for (int p = 0; p < 2; ++p) {   // B tiles: 16x128 each
            const int lin = tid + p * 256;
            const int kl = lin >> 5, ng = (lin & 31) << 2;
            *reinterpret_cast<float4*>(&Blr[0][kl][ng]) =
                *reinterpret_cast<const float4*>(br + (size_t)(k0 + kl) * NFFT + n0 + ng);
            *reinterpret_cast<float4*>(&Bli[0][kl][ng]) =
                *reinterpret_cast<const float4*>(bi + (size_t)(k0 + kl) * NFFT + n0 + ng);
        }
#endif
        __syncthreads();

        const int tcol = mw + (lane & 15);
#pragma unroll
        for (int kk = 0; kk < 16; kk += 4) {
            const int kr = kk + ((lane >> 4) << 1);
            v2f a_r, a_i;
            a_r[0] = Alr[buf][kr][tcol];  a_r[1] = Alr[buf][kr + 1][tcol];
            a_i[0] = Ali[buf][kr][tcol];  a_i[1] = Ali[buf][kr + 1][tcol];
#pragma unroll
            for (int nt = 0; nt < 4; ++nt) {
                const int nc = nw + (nt << 4) + (lane & 15);
                v2f b_r, b_i;
                b_r[0] = Blr[buf][kr][nc];  b_r[1] = Blr[buf][kr + 1][nc];
                b_i[0] = Bli[buf][kr][nc];  b_i[1] = Bli[buf][kr + 1][nc];
                acc[nt] = wmma_f32(a_r, b_r, acc[nt]);
                acc[nt] = wmma_f32(a_i, b_i, acc[nt]);     // ibuf pre-negated
            }
        }
        __syncthreads();
    }

    // overlap-add scatter: sig[t*HOP + n] += out[t,n]
    float* sc = sig + (size_t)bc * LP;
    const int tbase = t0 + mw + ((lane >> 4) << 3);
    const int nbase = n0 + nw + (lane & 15);
#pragma unroll
    for (int nt = 0; nt < 4; ++nt) {
        const int n = nbase + (nt << 4);
#pragma unroll
        for (int v = 0; v < 8; ++v) {
            const int t = tbase + v;
            if (t < NF)
                unsafeAtomicAdd(&sc[(size_t)t * HOP + n], acc[nt][v]);
        }
    }
}

// ---------------- kernel 4: Hann^2 normalization + slice -------------------
__global__ void norm_out(const float* __restrict__ sig, float* __restrict__ out) {
    const int bc = blockIdx.y;
    const int j  = blockIdx.x * blockDim.x + threadIdx.x;
    if (j >= T_LEN) return;
    const int g = j + OUT_OFF;
    const int d = g - (NFFT - 1);
    const int f_lo = (d > 0) ? ((d + HOP - 1) >> 10) : 0;
    int f_hi = g >> 10; if (f_hi > NF - 1) f_hi = NF - 1;
    float norm = 0.f;
    for (int f = f_lo; f <= f_hi; ++f) {
        const int n = g - (f << 10);
        const float wv = 0.5f * (1.0f - cosf((float)n * (6.28318530717958647692f / 4096.0f)));
        norm += wv * wv;
    }
    out[(size_t)bc * T_LEN + j] = sig[(size_t)bc * LP + g] / fmaxf(norm, 1e-8f);
}

// ---------------- host-side orchestration ----------------------------------
extern "C" void kernel_launch(void* const* d_in, const int* in_sizes, int n_in,
                              void* d_out, int out_size, void* d_ws, size_t ws_size,
                              hipStream_t stream) {
    (void)in_sizes; (void)n_in; (void)out_size; (void)ws_size;
    const float* mix = (const float*)d_in[0];
    const float* br  = (const float*)d_in[1];   // basis_real (2049 x 4096)
    const float* bi  = (const float*)d_in[2];   // basis_imag (2049 x 4096)
    float* out = (float*)d_out;

    float* xp   = (float*)d_ws;                         // BC*LP
    float* rbuf = xp   + (size_t)BC * LP;               // BC*NFREQ*NFP
    float* ibuf = rbuf + (size_t)BC * NFREQ * NFP;      // BC*NFREQ*NFP (negated)
    float* sig  = ibuf + (size_t)BC * NFREQ * NFP;      // BC*LP (overlap-add accum)

    const int n4 = BC * LP / 4;
    zero_kernel<<<(n4 + 255) / 256, 256, 0, stream>>>(sig, n4);
    pad_kernel<<<dim3((LP + 255) / 256, BC), 256, 0, stream>>>(mix, xp);
    stft_gemm<<<dim3(NFREQ / 128, NFP / 64, BC), 256, 0, stream>>>(xp, br, bi, rbuf, ibuf);
    istft_gemm_scatter<<<dim3(NFP / 64, NFFT / 128, BC), 256, 0, stream>>>(rbuf, ibuf, br, bi, sig);
    norm_out<<<dim3((T_LEN + 255) / 256, BC), 256, 0, stream>>>(sig, out);
}